// BiWindowMambaLayer_34093450396624
// MI455X (gfx1250) — compile-verified
//
#include <hip/hip_runtime.h>
#include <math.h>

// ---------------- problem constants ----------------
#define DIMC      192
#define D_STATE   16
#define D_CONV    4
#define D_INNER   384           // 2*192
#define DT_RANK   12
#define LSEQ      4096          // 16*16*16 pooled tokens
#define XD_LD     48            // padded row stride for x_dbl (44 real)
#define ZDIM      64

typedef __attribute__((ext_vector_type(2))) float v2f;
typedef __attribute__((ext_vector_type(4))) float v4f;
typedef __attribute__((ext_vector_type(8))) float v8f;

// ---------------- workspace layout (floats) ----------------
#define OFF_XF     0u                         // 4096*192
#define OFF_XN     (OFF_XF + LSEQ*DIMC)       // 4096*192
#define OFF_XZF    (OFF_XN + LSEQ*DIMC)       // 4096*768
#define OFF_XZB    (OFF_XZF + LSEQ*2*D_INNER)
#define OFF_XXF    (OFF_XZB + LSEQ*2*D_INNER) // 4096*384 (conv+silu)
#define OFF_XXB    (OFF_XXF + LSEQ*D_INNER)
#define OFF_XDF    (OFF_XXB + LSEQ*D_INNER)   // 4096*48
#define OFF_XDB    (OFF_XDF + LSEQ*XD_LD)
#define OFF_DTF    (OFF_XDB + LSEQ*XD_LD)     // 4096*384 softplus(dt)
#define OFF_DTB    (OFF_DTF + LSEQ*D_INNER)
#define OFF_YF     (OFF_DTB + LSEQ*D_INNER)   // 4096*384 scan output
#define OFF_YB     (OFF_YF + LSEQ*D_INNER)
#define OFF_S      (OFF_YB + LSEQ*D_INNER)    // 4096*384 combined
#define OFF_O      (OFF_S + LSEQ*D_INNER)     // 4096*192

__device__ __forceinline__ float sigmoidf_(float x) { return 1.0f / (1.0f + expf(-x)); }
__device__ __forceinline__ float softplusf_(float x) {
  return fmaxf(x, 0.0f) + log1pf(expf(-fabsf(x)));   // stable log(1+exp(x))
}

// ---------------- 1) 4x4x4 average pool: x[192][64][64][64] -> xf[l][c] ----------------
__global__ void pool_kernel(const float* __restrict__ x, float* __restrict__ xf) {
  int idx = blockIdx.x * blockDim.x + threadIdx.x;      // 192*4096 outputs
  if (idx >= DIMC * LSEQ) return;
  int c = idx >> 12;                                    // /4096
  int l = idx & 4095;
  int nz = l >> 8, nh = (l >> 4) & 15, nw = l & 15;
  const float* p = x + (size_t)c * (ZDIM*ZDIM*ZDIM) + (nz*4) * (ZDIM*ZDIM) + (nh*4) * ZDIM + nw*4;
  float s = 0.0f;
  #pragma unroll
  for (int dz = 0; dz < 4; ++dz)
    #pragma unroll
    for (int dh = 0; dh < 4; ++dh) {
      v4f v = *(const v4f*)(p + dz * (ZDIM*ZDIM) + dh * ZDIM);
      s += v.x + v.y + v.z + v.w;
    }
  xf[l * DIMC + c] = s * (1.0f / 64.0f);
}

// ---------------- 2) LayerNorm over C=192; one wave per token ----------------
__global__ void ln_kernel(const float* __restrict__ xf, const float* __restrict__ w,
                          const float* __restrict__ b, float* __restrict__ xn) {
  int wave = threadIdx.x >> 5;
  int lane = threadIdx.x & 31;
  int l = blockIdx.x * 8 + wave;                        // 8 waves per block
  const float* row = xf + (size_t)l * DIMC;
  float v[6], s = 0.0f, ss = 0.0f;
  #pragma unroll
  for (int i = 0; i < 6; ++i) { v[i] = row[lane * 6 + i]; s += v[i]; ss += v[i] * v[i]; }
  #pragma unroll
  for (int off = 16; off > 0; off >>= 1) { s += __shfl_xor(s, off, 32); ss += __shfl_xor(ss, off, 32); }
  float mu = s * (1.0f / DIMC);
  float var = ss * (1.0f / DIMC) - mu * mu;
  float rs = rsqrtf(var + 1e-5f);
  #pragma unroll
  for (int i = 0; i < 6; ++i) {
    int c = lane * 6 + i;
    xn[(size_t)l * DIMC + c] = (v[i] - mu) * rs * w[c] + b[c];
  }
}

// ---------------- 3) generic dual-branch WMMA f32 GEMM ----------------
// C = act(A(MxK) * W(NxK)^T [+bias]); one wave per 16x16 tile; blockIdx.z picks branch.
// epilogue: 0=none, 1=softplus(v+bias[n]).  flip: store row M-1-m (time reversal).
__global__ __launch_bounds__(32)
void gemm_wmma(const float* __restrict__ A0, const float* __restrict__ A1, int lda,
               const float* __restrict__ W0, const float* __restrict__ W1, int ldw, int K,
               float* __restrict__ C0, float* __restrict__ C1, int ldc, int M, int Nreal,
               const float* __restrict__ bias0, const float* __restrict__ bias1,
               int epilogue, int flip0, int flip1) {
  int br = blockIdx.z;
  const float* A    = br ? A1 : A0;
  const float* W    = br ? W1 : W0;
  float*       C    = br ? C1 : C0;
  const float* bias = br ? bias1 : bias0;
  int          flip = br ? flip1 : flip0;

  int lane = threadIdx.x;
  int half = lane >> 4;                 // 0 or 1
  int mi   = lane & 15;
  int mrow = blockIdx.x * 16 + mi;
  int ncol = blockIdx.y * 16 + mi;
  int ncl  = ncol < Nreal ? ncol : (Nreal - 1);   // clamp for safe loads
  int kh   = half << 1;                 // K sub-offset within a 16x16x4 step

  const float* Ap = A + (size_t)mrow * lda + kh;
  const float* Wp = W + (size_t)ncl  * ldw + kh;

  v8f acc = {};
  for (int k = 0; k < K; k += 4) {
    v2f a;  a.x  = Ap[k]; a.y  = Ap[k + 1];
    v2f bb; bb.x = Wp[k]; bb.y = Wp[k + 1];
    acc = __builtin_amdgcn_wmma_f32_16x16x4_f32(false, a, false, bb,
                                                (short)0, acc, false, false);
  }

  if (ncol < Nreal) {
    float bval = (epilogue == 1) ? bias[ncol] : 0.0f;
    #pragma unroll
    for (int r = 0; r < 8; ++r) {
      int ms = blockIdx.x * 16 + r + (half << 3);
      float v = acc[r];
      if (epilogue == 1) v = softplusf_(v + bval);
      int row = flip ? (M - 1 - ms) : ms;
      C[(size_t)row * ldc + ncol] = v;
    }
  }
}

// ---------------- 4) depthwise causal conv(4) + SiLU, both branches ----------------
__global__ void conv_silu_kernel(const float* __restrict__ xz0, const float* __restrict__ xz1,
                                 const float* __restrict__ cw0, const float* __restrict__ cw1,
                                 const float* __restrict__ cb0, const float* __restrict__ cb1,
                                 float* __restrict__ xxs0, float* __restrict__ xxs1) {
  int br = blockIdx.y;
  const float* xz = br ? xz1 : xz0;
  const float* cw = br ? cw1 : cw0;
  const float* cb = br ? cb1 : cb0;
  float*      xxs = br ? xxs1 : xxs0;

  int idx = blockIdx.x * blockDim.x + threadIdx.x;
  if (idx >= LSEQ * D_INNER) return;
  int t = idx / D_INNER;
  int d = idx - t * D_INNER;
  float acc = cb[d];
  #pragma unroll
  for (int j = 0; j < D_CONV; ++j) {
    int ts = t - 3 + j;
    if (ts >= 0) acc += cw[d * 4 + j] * xz[(size_t)ts * (2 * D_INNER) + d];
  }
  xxs[idx] = acc * sigmoidf_(acc);      // silu
}

// ---------------- 5) selective scan: 1 wave = 2 channels, lane%16 = state; both branches ----------------
__global__ __launch_bounds__(32)
void scan_kernel(const float* __restrict__ dt0, const float* __restrict__ dt1,
                 const float* __restrict__ xx0, const float* __restrict__ xx1,
                 const float* __restrict__ xd0, const float* __restrict__ xd1,
                 const float* __restrict__ al0, const float* __restrict__ al1,
                 float* __restrict__ y0, float* __restrict__ y1) {
  int br = blockIdx.y;
  const float* dt    = br ? dt1 : dt0;
  const float* xxs   = br ? xx1 : xx0;
  const float* xdbl  = br ? xd1 : xd0;
  const float* A_log = br ? al1 : al0;
  float*       y     = br ? y1 : y0;

  int lane = threadIdx.x;
  int half = lane >> 4;
  int n    = lane & 15;
  int d    = blockIdx.x * 2 + half;
  float Av = -expf(A_log[d * D_STATE + n]);
  float h = 0.0f;
  #pragma unroll 4
  for (int t = 0; t < LSEQ; ++t) {
    float dtv = dt[(size_t)t * D_INNER + d];
    float xv  = xxs[(size_t)t * D_INNER + d];
    float Bv  = xdbl[(size_t)t * XD_LD + DT_RANK + n];
    float Cv  = xdbl[(size_t)t * XD_LD + DT_RANK + D_STATE + n];
    float dA  = expf(dtv * Av);
    h = fmaf(dA, h, dtv * Bv * xv);
    float p = h * Cv;
    p += __shfl_xor(p, 1, 32);
    p += __shfl_xor(p, 2, 32);
    p += __shfl_xor(p, 4, 32);
    p += __shfl_xor(p, 8, 32);
    if (n == 0) y[(size_t)t * D_INNER + d] = p;
  }
}

// ---------------- 6) combine: S[l][d] = xm_f[l][d] + xm_b[L-1-l][d] ----------------
__global__ void combine_kernel(const float* __restrict__ yf, const float* __restrict__ xxf,
                               const float* __restrict__ xzf, const float* __restrict__ Df,
                               const float* __restrict__ yb, const float* __restrict__ xxb,
                               const float* __restrict__ xzb, const float* __restrict__ Db,
                               float* __restrict__ S) {
  int idx = blockIdx.x * blockDim.x + threadIdx.x;
  if (idx >= LSEQ * D_INNER) return;
  int l = idx / D_INNER;
  int d = idx - l * D_INNER;
  float zf = xzf[(size_t)l * (2 * D_INNER) + D_INNER + d];
  float xmf = (yf[idx] + Df[d] * xxf[idx]) * (zf * sigmoidf_(zf));
  int lb = LSEQ - 1 - l;
  size_t ib = (size_t)lb * D_INNER + d;
  float zb = xzb[(size_t)lb * (2 * D_INNER) + D_INNER + d];
  float xmb = (yb[ib] + Db[d] * xxb[ib]) * (zb * sigmoidf_(zb));
  S[idx] = xmf + xmb;
}

// ---------------- 7) 4x upsample + residual add (NT stores: output never re-read) ----------------
__global__ void upsample_kernel(const float* __restrict__ x, const float* __restrict__ o,
                                float* __restrict__ out) {
  int idx = blockIdx.x * blockDim.x + threadIdx.x;   // 192*64*64*16 float4 groups
  if (idx >= DIMC * ZDIM * ZDIM * 16) return;
  int wq = idx & 15;
  int hh = (idx >> 4) & 63;
  int zz = (idx >> 10) & 63;
  int c  = idx >> 16;
  int l  = (zz >> 2) * 256 + (hh >> 2) * 16 + wq;    // nw == wq (4 consecutive w share it)
  float val = o[(size_t)l * DIMC + c];
  size_t off = (size_t)idx * 4;
  v4f xv = *(const v4f*)(x + off);                   // RT: may hit lines left by pool
  v4f r = xv + val;                                  // scalar broadcast add
  __builtin_nontemporal_store(r, (v4f*)(out + off)); // NT: don't pollute L2
}

// ---------------- host launcher ----------------
extern "C" void kernel_launch(void* const* d_in, const int* in_sizes, int n_in,
                              void* d_out, int out_size, void* d_ws, size_t ws_size,
                              hipStream_t stream) {
  (void)in_sizes; (void)n_in; (void)out_size; (void)ws_size;
  const float* x      = (const float*)d_in[0];
  const float* ln_w   = (const float*)d_in[1];
  const float* ln_b   = (const float*)d_in[2];
  const float* out_w  = (const float*)d_in[3];
  const float* in_w_f   = (const float*)d_in[4];
  const float* conv_w_f = (const float*)d_in[5];
  const float* conv_b_f = (const float*)d_in[6];
  const float* xproj_f  = (const float*)d_in[7];
  const float* dt_w_f   = (const float*)d_in[8];
  const float* dt_b_f   = (const float*)d_in[9];
  const float* A_log_f  = (const float*)d_in[10];
  const float* D_f      = (const float*)d_in[11];
  const float* in_w_b   = (const float*)d_in[12];
  const float* conv_w_b = (const float*)d_in[13];
  const float* conv_b_b = (const float*)d_in[14];
  const float* xproj_b  = (const float*)d_in[15];
  const float* dt_w_b   = (const float*)d_in[16];
  const float* dt_b_b   = (const float*)d_in[17];
  const float* A_log_b  = (const float*)d_in[18];
  const float* D_b      = (const float*)d_in[19];

  float* ws  = (float*)d_ws;
  float* XF  = ws + OFF_XF;
  float* XN  = ws + OFF_XN;
  float* XZF = ws + OFF_XZF;
  float* XZB = ws + OFF_XZB;
  float* XXF = ws + OFF_XXF;
  float* XXB = ws + OFF_XXB;
  float* XDF = ws + OFF_XDF;
  float* XDB = ws + OFF_XDB;
  float* DTF = ws + OFF_DTF;
  float* DTB = ws + OFF_DTB;
  float* YF  = ws + OFF_YF;
  float* YB  = ws + OFF_YB;
  float* S   = ws + OFF_S;
  float* O   = ws + OFF_O;
  float* out = (float*)d_out;

  // 1) pool
  pool_kernel<<<dim3((DIMC * LSEQ + 255) / 256), dim3(256), 0, stream>>>(x, XF);
  // 2) layernorm
  ln_kernel<<<dim3(LSEQ / 8), dim3(256), 0, stream>>>(XF, ln_w, ln_b, XN);
  // 3) in_proj, both branches in one launch (backward stored time-reversed)
  gemm_wmma<<<dim3(LSEQ / 16, (2 * D_INNER) / 16, 2), dim3(32), 0, stream>>>(
      XN, XN, DIMC, in_w_f, in_w_b, DIMC, DIMC,
      XZF, XZB, 2 * D_INNER, LSEQ, 2 * D_INNER, nullptr, nullptr, 0, 0, 1);
  // 4) conv + silu, both branches
  conv_silu_kernel<<<dim3((LSEQ * D_INNER + 255) / 256, 2), dim3(256), 0, stream>>>(
      XZF, XZB, conv_w_f, conv_w_b, conv_b_f, conv_b_b, XXF, XXB);
  // 5) x_proj: (L x 384) x (44 x 384)^T -> L x 44 (ld 48), both branches
  gemm_wmma<<<dim3(LSEQ / 16, 3, 2), dim3(32), 0, stream>>>(
      XXF, XXB, D_INNER, xproj_f, xproj_b, D_INNER, D_INNER,
      XDF, XDB, XD_LD, LSEQ, DT_RANK + 2 * D_STATE, nullptr, nullptr, 0, 0, 0);
  // 6) dt_proj + softplus: (L x 12) x (384 x 12)^T -> L x 384, both branches
  gemm_wmma<<<dim3(LSEQ / 16, D_INNER / 16, 2), dim3(32), 0, stream>>>(
      XDF, XDB, XD_LD, dt_w_f, dt_w_b, DT_RANK, DT_RANK,
      DTF, DTB, D_INNER, LSEQ, D_INNER, dt_b_f, dt_b_b, 1, 0, 0);
  // 7) selective scan, both branches (384 concurrent waves)
  scan_kernel<<<dim3(D_INNER / 2, 2), dim3(32), 0, stream>>>(
      DTF, DTB, XXF, XXB, XDF, XDB, A_log_f, A_log_b, YF, YB);
  // 8) combine (+ D*x, * silu(z), flip backward)
  combine_kernel<<<dim3((LSEQ * D_INNER + 255) / 256), dim3(256), 0, stream>>>(
      YF, XXF, XZF, D_f, YB, XXB, XZB, D_b, S);
  // 9) out_proj: (L x 384) x (192 x 384)^T -> L x 192
  gemm_wmma<<<dim3(LSEQ / 16, DIMC / 16, 1), dim3(32), 0, stream>>>(
      S, S, D_INNER, out_w, out_w, D_INNER, D_INNER,
      O, O, DIMC, LSEQ, DIMC, nullptr, nullptr, 0, 0, 0);
  // 10) upsample + residual
  upsample_kernel<<<dim3((DIMC * ZDIM * ZDIM * 16 + 255) / 256), dim3(256), 0, stream>>>(x, O, out);
}